// SAGPool_46789373723272
// MI455X (gfx1250) — compile-verified
//
#include <hip/hip_runtime.h>
#include <hip/hip_bf16.h>
#include <math.h>

typedef float v2f __attribute__((ext_vector_type(2)));
typedef float v8f __attribute__((ext_vector_type(8)));

#define N_NODES 8192
#define D_FEAT  256
#define N_EDGES 262144
#define K_KEEP  4096
#define KSPLIT  2

// ---------------------------------------------------------------------------
// Kernel 1: xw[n] = sum_d x[n,d] * W[d]    (GEMV via V_WMMA_F32_16X16X4_F32)
// One wave32 handles 16 rows. A-operand: lane = row (mrow), half selects K pair.
// B-operand: vector in column 0 only (mrow==0 lanes), others masked to 0.0f
// so EXEC stays all-ones (WMMA requirement).
// ---------------------------------------------------------------------------
__global__ void sagpool_xw_wmma(const float* __restrict__ x,
                                const float* __restrict__ W,
                                float* __restrict__ xw) {
  const int lane  = threadIdx.x & 31;
  const int wave  = threadIdx.x >> 5;
  const int rowBase = (blockIdx.x * (blockDim.x >> 5) + wave) << 4;
  const int half  = lane >> 4;          // 0: K=0,1   1: K=2,3
  const int mrow  = lane & 15;          // row within 16-row tile
  const float bmask = (mrow == 0) ? 1.0f : 0.0f;
  const float* arow = x + (size_t)(rowBase + mrow) * D_FEAT + 2 * half;

  v8f c = {};
  for (int k = 0; k < D_FEAT; k += 4) {
    v2f a  = *(const v2f*)(arow + k);
    v2f wv = *(const v2f*)(W + k + 2 * half);
    v2f bb = { wv.x * bmask, wv.y * bmask };
    c = __builtin_amdgcn_wmma_f32_16x16x4_f32(false, a, false, bb,
                                              (short)0, c, false, false);
  }
  // C column 0: lane 0 -> M=0..7 in c[0..7]; lane 16 -> M=8..15.
  if (mrow == 0) {
    float* outp = xw + rowBase + 8 * half;
#pragma unroll
    for (int r = 0; r < 8; ++r) outp[r] = c[r];
  }
}

// ---------------------------------------------------------------------------
// Kernel 2: partial[ks][m] = sum_{k in slice ks} A[m,k] * xw[k]
// The 256MB A stream — the bandwidth-critical kernel. 16 rows per wave,
// K split in KSPLIT slices for occupancy, two accumulator chains for ILP.
// ---------------------------------------------------------------------------
__global__ void sagpool_score_wmma(const float* __restrict__ A,
                                   const float* __restrict__ xw,
                                   float* __restrict__ partial) {
  const int lane  = threadIdx.x & 31;
  const int wave  = threadIdx.x >> 5;
  const int rowBase = (blockIdx.x * (blockDim.x >> 5) + wave) << 4;
  const int ks    = blockIdx.y;
  const int kBeg  = ks * (N_NODES / KSPLIT);
  const int kEnd  = kBeg + (N_NODES / KSPLIT);
  const int half  = lane >> 4;
  const int mrow  = lane & 15;
  const float bmask = (mrow == 0) ? 1.0f : 0.0f;
  const float* arow = A + (size_t)(rowBase + mrow) * N_NODES + 2 * half;

  v8f c0 = {}, c1 = {};
  for (int k = kBeg; k < kEnd; k += 8) {
    __builtin_prefetch(arow + k + 128, 0, 1);   // global_prefetch ahead in row
    v2f a0 = *(const v2f*)(arow + k);
    v2f a1 = *(const v2f*)(arow + k + 4);
    v2f w0 = *(const v2f*)(xw + k + 2 * half);
    v2f w1 = *(const v2f*)(xw + k + 4 + 2 * half);
    v2f b0 = { w0.x * bmask, w0.y * bmask };
    v2f b1 = { w1.x * bmask, w1.y * bmask };
    c0 = __builtin_amdgcn_wmma_f32_16x16x4_f32(false, a0, false, b0,
                                               (short)0, c0, false, false);
    c1 = __builtin_amdgcn_wmma_f32_16x16x4_f32(false, a1, false, b1,
                                               (short)0, c1, false, false);
  }
  v8f c = c0 + c1;
  if (mrow == 0) {
    float* outp = partial + (size_t)ks * N_NODES + rowBase + 8 * half;
#pragma unroll
    for (int r = 0; r < 8; ++r) outp[r] = c[r];
  }
}

// score[m] = partial[0][m] + partial[1][m] + b   (fixed order -> deterministic)
__global__ void sagpool_score_finalize(const float* __restrict__ partial,
                                       const float* __restrict__ b,
                                       float* __restrict__ score) {
  int i = blockIdx.x * blockDim.x + threadIdx.x;
  if (i < N_NODES) score[i] = partial[i] + partial[N_NODES + i] + b[0];
}

// ---------------------------------------------------------------------------
// Kernel 3: exact top-K via single-workgroup bitonic sort in LDS (64KB).
// Descending by score, ties broken by ascending index (lax.top_k semantics).
// Also emits perm, top scores, node_map scatter, batch zeros.
// ---------------------------------------------------------------------------
__global__ void sagpool_topk(const float* __restrict__ score,
                             float* __restrict__ tsc,        // ws: K sorted scores
                             int* __restrict__ perm,         // ws: K
                             int* __restrict__ node_map,     // ws: N
                             int* __restrict__ perm_out,     // d_out section
                             long long* __restrict__ batch_out) {
  __shared__ float ss[N_NODES];
  __shared__ int   si[N_NODES];
  const int tid = threadIdx.x;
  for (int i = tid; i < N_NODES; i += blockDim.x) {
    ss[i] = score[i];
    si[i] = i;
    node_map[i] = -1;
  }
  __syncthreads();

  for (int k = 2; k <= N_NODES; k <<= 1) {
    for (int j = k >> 1; j > 0; j >>= 1) {
      for (int i = tid; i < N_NODES; i += blockDim.x) {
        int ixj = i ^ j;
        if (ixj > i) {
          float s1 = ss[i], s2 = ss[ixj];
          int   i1 = si[i], i2 = si[ixj];
          // "i1 precedes i2" in descending order with index tie-break
          bool prec = (s1 > s2) || (s1 == s2 && i1 < i2);
          bool descSeg = ((i & k) == 0);
          if (descSeg ? !prec : prec) {
            ss[i] = s2; ss[ixj] = s1;
            si[i] = i2; si[ixj] = i1;
          }
        }
      }
      __syncthreads();
    }
  }

  for (int i = tid; i < K_KEEP; i += blockDim.x) {
    float s = ss[i];
    int   p = si[i];
    tsc[i]      = s;
    perm[i]     = p;
    perm_out[i] = p;
    batch_out[i] = 0LL;
  }
  __syncthreads();
  for (int i = tid; i < K_KEEP; i += blockDim.x) {
    node_map[si[i]] = i;    // distinct targets, no race
  }
}

// ---------------------------------------------------------------------------
// Kernel 4: x_new[i,:] = x[perm[i],:] * tanh(top_score[i])   (float4 stream)
// ---------------------------------------------------------------------------
__global__ void sagpool_gather_scale(const float* __restrict__ x,
                                     const float* __restrict__ tsc,
                                     const int* __restrict__ perm,
                                     float* __restrict__ x_new) {
  const int row = blockIdx.x;           // K_KEEP rows
  const int d4  = threadIdx.x;          // 64 float4 per row
  const float s = tanhf(tsc[row]);
  const float4* src = (const float4*)(x + (size_t)perm[row] * D_FEAT);
  float4 v = src[d4];
  float4* dst = (float4*)(x_new + (size_t)row * D_FEAT);
  dst[d4] = make_float4(v.x * s, v.y * s, v.z * s, v.w * s);
}

// ---------------------------------------------------------------------------
// Kernel 5: edge filter/remap.
// ---------------------------------------------------------------------------
__global__ void sagpool_edges(const long long* __restrict__ ei,   // int64 [2,E]
                              const float* __restrict__ ea,
                              const int* __restrict__ node_map,
                              int* __restrict__ ei_new,           // int32 [2,E]
                              float* __restrict__ ea_new) {
  int e = blockIdx.x * blockDim.x + threadIdx.x;
  if (e >= N_EDGES) return;
  int r = node_map[(int)ei[e]];
  int c = node_map[(int)ei[N_EDGES + e]];
  bool valid = (r >= 0) && (c >= 0);
  ei_new[e]            = valid ? r : -1;
  ei_new[N_EDGES + e]  = valid ? c : -1;
  ea_new[e]            = valid ? ea[e] : 0.0f;
}

// ---------------------------------------------------------------------------
extern "C" void kernel_launch(void* const* d_in, const int* in_sizes, int n_in,
                              void* d_out, int out_size, void* d_ws, size_t ws_size,
                              hipStream_t stream) {
  const float*     x  = (const float*)d_in[0];     // [N, D]
  const float*     A  = (const float*)d_in[1];     // [N, N]
  const float*     W  = (const float*)d_in[2];     // [D, 1]
  const float*     b  = (const float*)d_in[3];     // [1]
  const long long* ei = (const long long*)d_in[4]; // [2, E] int64
  const float*     ea = (const float*)d_in[5];     // [E]

  // Workspace layout (bytes)
  char* ws = (char*)d_ws;
  float* xw       = (float*)(ws + 0);                         //  8192 f
  float* partial  = (float*)(ws + 32768);                     //  2*8192 f
  float* score    = (float*)(ws + 98304);                     //  8192 f
  float* tsc      = (float*)(ws + 131072);                    //  4096 f
  int*   perm     = (int*)  (ws + 147456);                    //  4096 i32
  int*   node_map = (int*)  (ws + 163840);                    //  8192 i32

  // Output layout (bytes), reference return order:
  // x_new f32[K,D] | edge_index_new i32[2,E] | edge_attr_new f32[E]
  // | batch_new i64[K] | perm i32[K]
  char* out = (char*)d_out;
  size_t off = 0;
  float*     x_new     = (float*)(out + off);  off += (size_t)K_KEEP * D_FEAT * 4;
  int*       ei_new    = (int*)  (out + off);  off += (size_t)2 * N_EDGES * 4;
  float*     ea_new    = (float*)(out + off);  off += (size_t)N_EDGES * 4;
  long long* batch_new = (long long*)(out + off); off += (size_t)K_KEEP * 8;
  int*       perm_out  = (int*)  (out + off);

  // 1) xw = x @ W : 512 row-tiles, 8 waves/block -> 64 blocks
  sagpool_xw_wmma<<<dim3(N_NODES / (16 * 8)), dim3(256), 0, stream>>>(x, W, xw);

  // 2) partial = A @ xw (K-split): 2 waves/block, 32 rows/block, grid 256 x 2
  sagpool_score_wmma<<<dim3(N_NODES / (16 * 2), KSPLIT), dim3(64), 0, stream>>>(
      A, xw, partial);

  // 3) score = sum(partial) + b
  sagpool_score_finalize<<<dim3(N_NODES / 256), dim3(256), 0, stream>>>(
      partial, b, score);

  // 4) top-K bitonic sort + perm/node_map/batch
  sagpool_topk<<<dim3(1), dim3(1024), 0, stream>>>(
      score, tsc, perm, node_map, perm_out, batch_new);

  // 5) x_new gather * tanh(score)
  sagpool_gather_scale<<<dim3(K_KEEP), dim3(D_FEAT / 4), 0, stream>>>(
      x, tsc, perm, x_new);

  // 6) edge remap
  sagpool_edges<<<dim3(N_EDGES / 256), dim3(256), 0, stream>>>(
      ei, ea, node_map, ei_new, ea_new);
}